// DeepMMD_48541720380025
// MI455X (gfx1250) — compile-verified
//
#include <hip/hip_runtime.h>

typedef __attribute__((ext_vector_type(2))) float v2f;
typedef __attribute__((ext_vector_type(8))) float v8f;

#define D_DIM   640
#define S_DIM   25
#define NPROTO  64
#define NQUERY  1024
#define NPAD    32
#define PCHUNK  8
#define NCOLS   (PCHUNK * NPAD)     // 256
#define KCH     64
#define PTCOLS  (NPROTO * NPAD)     // 2048

// workspace offsets (floats) — total ~2.05M floats (~8.2 MB)
#define WS_PT    0
#define WS_MUQ   (WS_PT   + D_DIM * PTCOLS)
#define WS_Q2    (WS_MUQ  + NQUERY * S_DIM)
#define WS_DQQ   (WS_Q2   + NQUERY * S_DIM)
#define WS_SDQQ  (WS_DQQ  + NQUERY * S_DIM * S_DIM)
#define WS_DPP   (WS_SDQQ + NQUERY)
#define WS_SDPP  (WS_DPP  + NPROTO * S_DIM * S_DIM)
#define WS_P2    (WS_SDPP + NPROTO)

__global__ __launch_bounds__(256) void mmd_prep_proto(const float* __restrict__ proto,
                                                      float* __restrict__ ws) {
  __shared__ float sm[D_DIM * S_DIM];
  __shared__ float mu[S_DIM];
  __shared__ float ssum;
  const int p = blockIdx.x;
  const int tid = threadIdx.x;
  const float* src = proto + p * (D_DIM * S_DIM);
  for (int i = tid; i < D_DIM * S_DIM; i += 256) sm[i] = src[i];
  if (tid == 0) ssum = 0.0f;
  __syncthreads();
  if (tid < S_DIM) {
    float acc = 0.0f;
    for (int d = 0; d < D_DIM; ++d) acc += sm[d * S_DIM + tid];
    mu[tid] = acc * (1.0f / D_DIM);
  }
  __syncthreads();
  for (int i = tid; i < D_DIM * S_DIM; i += 256) sm[i] -= mu[i % S_DIM];
  __syncthreads();
  // write B matrix: PT[d][p*32 + j], j>=25 zero-padded
  float* pt = ws + WS_PT;
  for (int i = tid; i < D_DIM * NPAD; i += 256) {
    int d = i >> 5, j = i & 31;
    float v = (j < S_DIM) ? sm[d * S_DIM + j] : 0.0f;
    pt[d * PTCOLS + p * NPAD + j] = v;
  }
  if (tid < S_DIM) {
    float acc = 0.0f;
    for (int d = 0; d < D_DIM; ++d) { float x = sm[d * S_DIM + tid]; acc += x * x; }
    ws[WS_P2 + p * S_DIM + tid] = acc;
  }
  float part = 0.0f;
  for (int e = tid; e < S_DIM * S_DIM; e += 256) {
    int i = e / S_DIM, j = e % S_DIM;
    float acc = 0.0f;
    for (int d = 0; d < D_DIM; ++d) {
      float diff = sm[d * S_DIM + i] - sm[d * S_DIM + j];
      acc += diff * diff;
    }
    ws[WS_DPP + p * S_DIM * S_DIM + e] = acc;
    part += acc;
  }
  atomicAdd(&ssum, part);
  __syncthreads();
  if (tid == 0) ws[WS_SDPP + p] = ssum;
}

__global__ __launch_bounds__(256) void mmd_prep_query(const float* __restrict__ query,
                                                      float* __restrict__ ws) {
  __shared__ float sm[D_DIM * S_DIM];
  __shared__ float mu[S_DIM];
  __shared__ float ssum;
  const int qi = blockIdx.x;
  const int tid = threadIdx.x;
  const float* src = query + qi * (D_DIM * S_DIM);
  for (int i = tid; i < D_DIM * S_DIM; i += 256) sm[i] = src[i];
  if (tid == 0) ssum = 0.0f;
  __syncthreads();
  if (tid < S_DIM) {
    float acc = 0.0f;
    for (int d = 0; d < D_DIM; ++d) acc += sm[d * S_DIM + tid];
    float m = acc * (1.0f / D_DIM);
    mu[tid] = m;
    ws[WS_MUQ + qi * S_DIM + tid] = m;
  }
  __syncthreads();
  for (int i = tid; i < D_DIM * S_DIM; i += 256) sm[i] -= mu[i % S_DIM];
  __syncthreads();
  if (tid < S_DIM) {
    float acc = 0.0f;
    for (int d = 0; d < D_DIM; ++d) { float x = sm[d * S_DIM + tid]; acc += x * x; }
    ws[WS_Q2 + qi * S_DIM + tid] = acc;
  }
  float part = 0.0f;
  for (int e = tid; e < S_DIM * S_DIM; e += 256) {
    int i = e / S_DIM, j = e % S_DIM;
    float acc = 0.0f;
    for (int d = 0; d < D_DIM; ++d) {
      float diff = sm[d * S_DIM + i] - sm[d * S_DIM + j];
      acc += diff * diff;
    }
    ws[WS_DQQ + qi * S_DIM * S_DIM + e] = acc;
    part += acc;
  }
  atomicAdd(&ssum, part);
  __syncthreads();
  if (tid == 0) ws[WS_SDQQ + qi] = ssum;
}

// One block = (query q, 8 protos). 8 waves; each wave owns a 2Mx2N block of
// 16x16 tiles (4 accumulators) -> 4 LDS b64 reads per 4 WMMAs per K-step.
__global__ __launch_bounds__(256) void mmd_main(const float* __restrict__ query,
                                                const float* __restrict__ ws,
                                                float* __restrict__ out) {
  __shared__ float a_lds[32 * KCH];        // 8 KB, A chunk row-major [m][kk]
  __shared__ float b_lds[KCH * NCOLS];     // 64 KB, pair-interleaved; reused as gram[32][256]
  __shared__ float mus[S_DIM];
  __shared__ float q2s[S_DIM];
  __shared__ float p2s[PCHUNK * S_DIM];
  __shared__ float red[2 * PCHUNK];

  const int q   = blockIdx.x >> 3;
  const int pc  = blockIdx.x & 7;
  const int p0  = pc * PCHUNK;
  const int tid = threadIdx.x;
  const int lane = tid & 31;
  const int w    = tid >> 5;               // wave id: owns N-tiles 2w, 2w+1
  const float* pt = ws + WS_PT;
  const float* qbase = query + q * (D_DIM * S_DIM);

  if (tid < S_DIM) { mus[tid] = ws[WS_MUQ + q * S_DIM + tid];
                     q2s[tid] = ws[WS_Q2  + q * S_DIM + tid]; }
  if (tid < PCHUNK * S_DIM) p2s[tid] = ws[WS_P2 + p0 * S_DIM + tid];
  if (tid < 2 * PCHUNK) red[tid] = 0.0f;
  __syncthreads();

  v8f acc00 = {};   // (M-tile 0, N-tile 2w)
  v8f acc01 = {};   // (M-tile 0, N-tile 2w+1)
  v8f acc10 = {};   // (M-tile 1, N-tile 2w)
  v8f acc11 = {};   // (M-tile 1, N-tile 2w+1)

  for (int kc = 0; kc < D_DIM / KCH; ++kc) {
    // stage A: centered query rows (rows >=25 are zero)
    for (int i = tid; i < 32 * KCH; i += 256) {
      int m = i & 31, kk = i >> 5;
      float v = 0.0f;
      if (m < S_DIM) v = qbase[(kc * KCH + kk) * S_DIM + m] - mus[m];
      a_lds[m * KCH + kk] = v;
    }
    // stage B: pair-interleaved so each lane's (K, K+1) fragment is one b64 read
    for (int i = tid; i < KCH * NCOLS; i += 256) {
      int c = i & (NCOLS - 1), kk = i >> 8;
      float v = pt[(kc * KCH + kk) * PTCOLS + pc * NCOLS + c];
      b_lds[(kk >> 1) * (2 * NCOLS) + c * 2 + (kk & 1)] = v;
    }
    // prefetch next K-chunk (global_prefetch_b8) while this chunk computes
    if (kc + 1 < D_DIM / KCH) {
      __builtin_prefetch(&pt[((kc + 1) * KCH + (tid >> 2)) * PTCOLS + pc * NCOLS
                             + ((tid & 3) << 6)], 0, 0);
      if (tid < KCH) __builtin_prefetch(&qbase[((kc + 1) * KCH + tid) * S_DIM], 0, 0);
    }
    __syncthreads();

    const int row0  = lane & 15;
    const int khalf = lane >> 4;           // lanes 16-31 hold K+2,K+3
    const int nb0   = (w * 32 + row0) * 2; // N-tile 2w column (pair-interleaved)
    #pragma unroll 4
    for (int ks = 0; ks < KCH / 4; ++ks) {
      const int k0 = ks * 4;
      const int brow = ((k0 >> 1) + khalf) * (2 * NCOLS);
      v2f a0 = *(const v2f*)&a_lds[row0 * KCH + k0 + 2 * khalf];
      v2f a1 = *(const v2f*)&a_lds[(row0 + 16) * KCH + k0 + 2 * khalf];
      v2f b0 = *(const v2f*)&b_lds[brow + nb0];
      v2f b1 = *(const v2f*)&b_lds[brow + nb0 + 32];
      acc00 = __builtin_amdgcn_wmma_f32_16x16x4_f32(false, a0, false, b0, (short)0, acc00, false, false);
      acc10 = __builtin_amdgcn_wmma_f32_16x16x4_f32(false, a1, false, b0, (short)0, acc10, false, false);
      acc01 = __builtin_amdgcn_wmma_f32_16x16x4_f32(false, a0, false, b1, (short)0, acc01, false, false);
      acc11 = __builtin_amdgcn_wmma_f32_16x16x4_f32(false, a1, false, b1, (short)0, acc11, false, false);
    }
    __syncthreads();
  }

  // spill Gram to LDS (C layout: VGPR r -> M=r (lanes 0-15), M=r+8 (lanes 16-31))
  float* gram = b_lds;                     // 32 x 256
  {
    const int n0 = w * 32 + (lane & 15);
    const int mh = (lane >> 4) * 8;
    #pragma unroll
    for (int r = 0; r < 8; ++r) {
      gram[(mh + r) * NCOLS + n0]           = acc00[r];
      gram[(mh + r) * NCOLS + n0 + 16]      = acc01[r];
      gram[(mh + r + 16) * NCOLS + n0]      = acc10[r];
      gram[(mh + r + 16) * NCOLS + n0 + 16] = acc11[r];
    }
  }
  __syncthreads();

  // ---- exp / MMD phase: 1 wave per proto ----
  const int pl  = tid >> 5;                // 0..7
  const int t32 = tid & 31;
  const int p   = p0 + pl;

  float part = 0.0f;                       // sum of dqp for bandwidth
  for (int e = t32; e < S_DIM * S_DIM; e += 32) {
    int i = e / S_DIM, j = e % S_DIM;
    part += q2s[i] + p2s[pl * S_DIM + j] - 2.0f * gram[i * NCOLS + pl * NPAD + j];
  }
  atomicAdd(&red[pl], part);
  __syncthreads();
  const float sdqp = red[pl];
  // bw = (sum_dqq + sum_dpp + 2*sum_dqp) / (50^2 - 50) / KERNEL_MUL^(5//2)
  const float bw   = (ws[WS_SDQQ + q] + ws[WS_SDPP + p] + 2.0f * sdqp)
                     * (1.0f / 2450.0f) * 0.25f;
  const float inv0 = 1.0f / bw;

  const float* dqq = ws + WS_DQQ + q * (S_DIM * S_DIM);
  const float* dpp = ws + WS_DPP + p * (S_DIM * S_DIM);
  float accum = 0.0f;
  for (int e = t32; e < S_DIM * S_DIM; e += 32) {
    int i = e / S_DIM, j = e % S_DIM;
    float vqq = dqq[e];
    float vpp = dpp[e];
    float vqp = q2s[i] + p2s[pl * S_DIM + j] - 2.0f * gram[i * NCOLS + pl * NPAD + j];
    float ik = inv0;
    #pragma unroll
    for (int k = 0; k < 5; ++k) {
      accum += __expf(-vqq * ik) + __expf(-vpp * ik) - 2.0f * __expf(-vqp * ik);
      ik *= 0.5f;                          // inv_k = inv0 / KERNEL_MUL^k
    }
  }
  atomicAdd(&red[PCHUNK + pl], accum);
  __syncthreads();
  if (t32 == 0) out[q * NPROTO + p] = -red[PCHUNK + pl] * (1.0f / 625.0f);
}

extern "C" void kernel_launch(void* const* d_in, const int* in_sizes, int n_in,
                              void* d_out, int out_size, void* d_ws, size_t ws_size,
                              hipStream_t stream) {
  (void)in_sizes; (void)n_in; (void)out_size; (void)ws_size;
  const float* proto = (const float*)d_in[0];
  const float* query = (const float*)d_in[1];
  float* out = (float*)d_out;
  float* ws  = (float*)d_ws;
  hipLaunchKernelGGL(mmd_prep_proto, dim3(NPROTO), dim3(256), 0, stream, proto, ws);
  hipLaunchKernelGGL(mmd_prep_query, dim3(NQUERY), dim3(256), 0, stream, query, ws);
  hipLaunchKernelGGL(mmd_main, dim3(NQUERY * PCHUNK), dim3(256), 0, stream, query, ws, out);
}